// Net_67035849555969
// MI455X (gfx1250) — compile-verified
//
#include <hip/hip_runtime.h>
#include <hip/hip_bf16.h>
#include <math.h>

#define NEGV (-1e30f)

typedef __attribute__((ext_vector_type(2))) float v2f;
typedef __attribute__((ext_vector_type(8))) float v8f;

// ---- monotonic float<->u32 key (order-preserving for atomicMax) ----
__device__ __forceinline__ unsigned fkey(float f) {
    unsigned u = __float_as_uint(f);
    return (u & 0x80000000u) ? ~u : (u | 0x80000000u);
}
__device__ __forceinline__ float funkey(unsigned k) {
    unsigned u = (k & 0x80000000u) ? (k & 0x7fffffffu) : ~k;
    return __uint_as_float(u);
}
static inline unsigned h_fkey(float f) {
    union { float f; unsigned u; } x; x.f = f;
    return (x.u & 0x80000000u) ? ~x.u : (x.u | 0x80000000u);
}

// ---- generic fill ----
__global__ void k_fill_u32(unsigned* __restrict__ p, unsigned v, long n) {
    long t = (long)blockIdx.x * blockDim.x + threadIdx.x;
    if (t < n) p[t] = v;
}

// ---- fused spline-conv edge pass ----
// block: (FOUT, 4); one edge per threadIdx.y, one output feature per threadIdx.x
template<int FIN, int FOUT>
__global__ __launch_bounds__(FOUT * 4)
void k_spline_edges(const float* __restrict__ x,
                    const int* __restrict__ src,
                    const int* __restrict__ dst,
                    const float* __restrict__ pseudo,
                    const float* __restrict__ W,   // [125, FIN, FOUT]
                    float* __restrict__ agg,       // [N, FOUT]
                    float* __restrict__ deg,       // [N]
                    int E)
{
    constexpr int EPB = 4;
    __shared__ float xs[EPB * FIN];
    __shared__ float bs[EPB * 8];
    __shared__ int   ks[EPB * 8];
    __shared__ int   sd[EPB * 2];

    const int ey = threadIdx.y;
    const long e = (long)blockIdx.x * EPB + ey;
    const bool valid = e < E;

    if (valid && threadIdx.x == 0) {
        int s = src[e];
        sd[ey * 2 + 0] = s;
        sd[ey * 2 + 1] = dst[e];
        float f[3]; int i0[3];
#pragma unroll
        for (int d = 0; d < 3; ++d) {
            float p = pseudo[e * 3 + d];
            p = fminf(fmaxf(p, 0.f), 1.f) * 4.0f;   // (K-1)=4
            float fl = fminf(floorf(p), 3.0f);      // clip to K-2
            i0[d] = (int)fl;
            f[d] = p - fl;
        }
#pragma unroll
        for (int j = 0; j < 8; ++j) {
            int a = (j >> 2) & 1, b = (j >> 1) & 1, c = j & 1;
            bs[ey * 8 + j] = (a ? f[0] : 1.f - f[0]) *
                             (b ? f[1] : 1.f - f[1]) *
                             (c ? f[2] : 1.f - f[2]);
            ks[ey * 8 + j] = (i0[0] + a) + (i0[1] + b) * 5 + (i0[2] + c) * 25;
        }
        __builtin_prefetch(W + (long)ks[ey * 8 + 0] * FIN * FOUT, 0, 0);
    }
    __syncthreads();
    if (valid) {
        int s = sd[ey * 2];
        for (int i = threadIdx.x; i < FIN; i += blockDim.x)
            xs[ey * FIN + i] = x[(long)s * FIN + i];
    }
    __syncthreads();
    if (!valid) return;

    const int o = threadIdx.x;
    int kj[8];
#pragma unroll
    for (int j = 0; j < 8; ++j) kj[j] = ks[ey * 8 + j];

    constexpr int CH = (FIN < 16) ? FIN : 16;   // x register-cache chunk
    float sj[8] = {0.f, 0.f, 0.f, 0.f, 0.f, 0.f, 0.f, 0.f};
    for (int i0 = 0; i0 < FIN; i0 += CH) {
        float xr[CH];
#pragma unroll
        for (int ii = 0; ii < CH; ++ii) xr[ii] = xs[ey * FIN + i0 + ii];
#pragma unroll
        for (int j = 0; j < 8; ++j) {
            const float* Wk = W + ((long)kj[j] * FIN + i0) * FOUT + o;
#pragma unroll
            for (int ii = 0; ii < CH; ++ii)
                sj[j] = fmaf(xr[ii], Wk[(long)ii * FOUT], sj[j]);
        }
    }
    float acc = 0.f;
#pragma unroll
    for (int j = 0; j < 8; ++j) acc = fmaf(bs[ey * 8 + j], sj[j], acc);

    int dn = sd[ey * 2 + 1];
    atomicAdd(&agg[(long)dn * FOUT + o], acc);
    if (o == 0) atomicAdd(&deg[dn], 1.0f);
}

// ---- node update: h = act(agg/max(deg,1) + x@root + bias), x@root via fp32 WMMA ----
// one wave per 16x16 output tile; x rows padded to 16-multiples (pad rows are zero)
template<int FIN, int FOUT>
__global__ __launch_bounds__(32)
void k_node_update(const float* __restrict__ x,    // [Npad,FIN]
                   const float* __restrict__ agg,  // [N,FOUT]
                   const float* __restrict__ deg,  // [N]
                   const float* __restrict__ root, // [FIN,FOUT]
                   const float* __restrict__ bias, // [FOUT]
                   float* __restrict__ h,          // [N,FOUT]
                   int N, int act)
{
    const int lane = threadIdx.x;      // 0..31 (wave32)
    const int half = lane >> 4;
    const int l16  = lane & 15;
    const int row0 = blockIdx.x * 16;
    const int col0 = blockIdx.y * 16;
    const int r    = row0 + l16;
    const int nb   = col0 + l16;

    v8f c = {};
    constexpr int KC = (FIN + 3) & ~3;
#pragma unroll
    for (int kk = 0; kk < KC; kk += 4) {
        const int ka = kk + 2 * half;
        v2f a, b;
        if constexpr (FIN % 4 == 0) {
            a = *(const v2f*)(x + (long)r * FIN + ka);            // 8B vector load
            b.x = root[(long)ka * FOUT + nb];
            b.y = root[(long)(ka + 1) * FOUT + nb];
        } else {  // FIN==1 rank-1 case, compile-time folded selects
            a.x = (ka < FIN) ? x[(long)r * FIN + ka] : 0.f;
            a.y = 0.f;
            b.x = (ka < FIN) ? root[(long)ka * FOUT + nb] : 0.f;
            b.y = 0.f;
        }
        c = __builtin_amdgcn_wmma_f32_16x16x4_f32(
                /*neg_a=*/false, a, /*neg_b=*/false, b,
                /*c_mod=*/(short)0, c, /*reuse_a=*/false, /*reuse_b=*/false);
    }

    const float bcol = bias[nb];
#pragma unroll
    for (int rr = 0; rr < 8; ++rr) {
        int row = row0 + rr + 8 * half;
        if (row >= N) continue;
        float d = fmaxf(deg[row], 1.0f);
        float v = agg[(long)row * FOUT + nb] / d + c[rr] + bcol;
        v = act ? fmaxf(v, 0.f) : (v > 0.f ? v : (expf(v) - 1.f));
        h[(long)row * FOUT + nb] = v;
    }
}

// ---- attention weight per node + segment max of weight ----
__global__ void k_att(const float* __restrict__ h, const float* __restrict__ attw,
                      const float* __restrict__ attb, const int* __restrict__ cluster,
                      float* __restrict__ wbuf, unsigned* __restrict__ segw,
                      int N, int F)
{
    int n = blockIdx.x * blockDim.x + threadIdx.x;
    if (n >= N) return;
    float acc = attb[0];
#pragma unroll 8
    for (int i = 0; i < F; ++i) acc = fmaf(h[(long)n * F + i], attw[i * 2], acc);
    wbuf[n] = acc;
    atomicMax(&segw[cluster[n]], fkey(acc));
}

// ---- scatter-max pool (weighted: only per-cluster att winners contribute) ----
__global__ void k_pool_scatter(const float* __restrict__ h, const float* __restrict__ wbuf,
                               const unsigned* __restrict__ segw, const int* __restrict__ cluster,
                               unsigned* __restrict__ pooled, int N, int F, int weighted)
{
    long t = (long)blockIdx.x * blockDim.x + threadIdx.x;
    if (t >= (long)N * F) return;
    int n = (int)(t / F), f = (int)(t % F);
    int cl = cluster[n];
    bool win = true;
    if (weighted) win = wbuf[n] >= funkey(segw[cl]);
    float v = win ? h[(long)n * F + f] : NEGV;
    atomicMax(&pooled[(long)cl * F + f], fkey(v));
}

__global__ void k_pool_finalize(const unsigned* __restrict__ pooled,
                                float* __restrict__ out, long n)
{
    long t = (long)blockIdx.x * blockDim.x + threadIdx.x;
    if (t >= n) return;
    float v = funkey(pooled[t]);
    out[t] = (v <= NEGV * 0.5f) ? 0.f : v;
}

// ---- FC head: z[16,512] @ fcw[512,10] + fcb, log_softmax. One wave, fp32 WMMA ----
__global__ __launch_bounds__(32)
void k_fc_head(const float* __restrict__ z,
               const float* __restrict__ fcw,
               const float* __restrict__ fcb,
               float* __restrict__ out)
{
    __shared__ float wpad[512][16];   // fcw zero-padded to 16 cols (32 KB LDS)
    __shared__ float lg[16][16];
    const int lane = threadIdx.x;
    const int half = lane >> 4, l16 = lane & 15;

    for (int t = lane; t < 512 * 16; t += 32) {
        int row = t >> 4, col = t & 15;
        wpad[row][col] = (col < 10) ? fcw[row * 10 + col] : 0.f;
    }
    __syncthreads();

    v8f c = {};
    for (int kk = 0; kk < 512; kk += 4) {
        int ka = kk + 2 * half;
        v2f a = *(const v2f*)(z + l16 * 512 + ka);
        v2f b;
        b.x = wpad[ka][l16];
        b.y = wpad[ka + 1][l16];
        c = __builtin_amdgcn_wmma_f32_16x16x4_f32(false, a, false, b, (short)0, c, false, false);
    }
#pragma unroll
    for (int rr = 0; rr < 8; ++rr)
        lg[rr + 8 * half][l16] = c[rr];
    __syncthreads();
    if (lane < 16) {
        float fb[10]; float m = -3.4e38f;
#pragma unroll
        for (int j = 0; j < 10; ++j) { fb[j] = lg[lane][j] + fcb[j]; m = fmaxf(m, fb[j]); }
        float s = 0.f;
#pragma unroll
        for (int j = 0; j < 10; ++j) s += expf(fb[j] - m);
        float lse = m + logf(s);
#pragma unroll
        for (int j = 0; j < 10; ++j) out[lane * 10 + j] = fb[j] - lse;
    }
}

static void fillu(unsigned* p, unsigned v, long n, hipStream_t stream) {
    if (n <= 0) return;
    long g = (n + 255) / 256;
    k_fill_u32<<<dim3((unsigned)g), dim3(256), 0, stream>>>(p, v, n);
}

template<int FIN, int FOUT>
static void launch_layer(const float* xin, const int* src, const int* dst,
                         const float* pseudo, const float* Wl, const float* rootl,
                         const float* bl, float* agg, float* deg, float* h,
                         int N, int E, int act, hipStream_t stream)
{
    dim3 eb(FOUT, 4);
    int eg = (E + 3) / 4;
    k_spline_edges<FIN, FOUT><<<dim3(eg), eb, 0, stream>>>(
        xin, src, dst, pseudo, Wl, agg, deg, E);
    dim3 ng((N + 15) / 16, FOUT / 16);
    k_node_update<FIN, FOUT><<<ng, 32, 0, stream>>>(
        xin, agg, deg, rootl, bl, h, N, act);
}

extern "C" void kernel_launch(void* const* d_in, const int* in_sizes, int n_in,
                              void* d_out, int out_size, void* d_ws, size_t ws_size,
                              hipStream_t stream) {
    (void)in_sizes; (void)n_in; (void)out_size; (void)ws_size;

    const int Nn[5] = {30000, 10000, 4000, 1500, 600};
    const int Ee[5] = {360000, 120000, 48000, 18000, 7200};
    const int Fo[5] = {32, 64, 64, 64, 64};
    const int Cc[5] = {10000, 4000, 1500, 600, 128};
    int Cp[5];  // cluster counts padded to 16 rows (next layer's WMMA row tiles)
    for (int l = 0; l < 5; ++l) Cp[l] = (Cc[l] + 15) & ~15;

    // ---- workspace carving (floats / u32, 8B alignment preserved) ----
    float* base = (float*)d_ws;
    float* W_agg = base;                    base += 30000 * 32;   // max N*Fout
    float* W_deg = base;                    base += 30000;
    float* W_att = base;                    base += 30000;
    unsigned* W_segw  = (unsigned*)base;    base += 10000;
    unsigned* W_poolk = (unsigned*)base;    base += 10000 * 32;   // max C*Fout
    float* hb[5]; float* pb[5];
    for (int l = 0; l < 5; ++l) {
        hb[l] = base; base += (long)Nn[l] * Fo[l];
        pb[l] = base; base += (long)Cp[l] * Fo[l];   // padded rows
    }

    const unsigned negk = h_fkey(NEGV);

    const float* xin = (const float*)d_in[0];
    for (int l = 0; l < 5; ++l) {
        const int*   src    = (const int*)d_in[1 + 3 * l];
        const int*   dst    = (const int*)d_in[2 + 3 * l];
        const float* pseudo = (const float*)d_in[3 + 3 * l];
        const int*   clus   = (const int*)d_in[16 + l];
        const float* Wl     = (const float*)d_in[21 + 3 * l];
        const float* rootl  = (const float*)d_in[22 + 3 * l];
        const float* bl     = (const float*)d_in[23 + 3 * l];
        const int N = Nn[l], E = Ee[l], Fout = Fo[l], C = Cc[l];

        fillu((unsigned*)W_agg, 0u, (long)N * Fout, stream);
        fillu((unsigned*)W_deg, 0u, N, stream);

        const int act = (l == 4) ? 1 : 0;
        if (l == 0)
            launch_layer<1, 32>(xin, src, dst, pseudo, Wl, rootl, bl,
                                W_agg, W_deg, hb[l], N, E, act, stream);
        else if (l == 1)
            launch_layer<32, 64>(xin, src, dst, pseudo, Wl, rootl, bl,
                                 W_agg, W_deg, hb[l], N, E, act, stream);
        else
            launch_layer<64, 64>(xin, src, dst, pseudo, Wl, rootl, bl,
                                 W_agg, W_deg, hb[l], N, E, act, stream);

        if (l < 4) {
            const float* attw = (const float*)d_in[36 + 2 * l];
            const float* attb = (const float*)d_in[37 + 2 * l];
            fillu(W_segw, negk, C, stream);
            k_att<<<(N + 255) / 256, 256, 0, stream>>>(
                hb[l], attw, attb, clus, W_att, W_segw, N, Fout);
        }

        fillu(W_poolk, negk, (long)C * Fout, stream);
        long tot = (long)N * Fout;
        k_pool_scatter<<<(unsigned)((tot + 255) / 256), 256, 0, stream>>>(
            hb[l], W_att, W_segw, clus, W_poolk, N, Fout, (l < 4) ? 1 : 0);
        long ctot = (long)C * Fout;
        k_pool_finalize<<<(unsigned)((ctot + 255) / 256), 256, 0, stream>>>(
            W_poolk, pb[l], ctot);
        // zero the 16-row padding tail so next layer's WMMA loads read zeros
        fillu((unsigned*)(pb[l] + (long)C * Fout), 0u,
              (long)(Cp[l] - C) * Fout, stream);

        xin = pb[l];
    }

    // pb[4] is [128,64] row-major == z[16,512] reshaped view
    k_fc_head<<<1, 32, 0, stream>>>(
        pb[4], (const float*)d_in[44], (const float*)d_in[45], (float*)d_out);
}